// CRF_56255481643046
// MI455X (gfx1250) — compile-verified
//
#include <hip/hip_runtime.h>
#include <math.h>

// CRF loss on gfx1250 (MI455X).
//
// Forward recursion rewritten in exp-space:
//   new[j] = m_b + log( sum_i exp(part[i]-m_b) * E[i,j] ),  E = exp(trans)
// E is a constant 50x50 f32 matrix (masked -1e4 entries underflow to exact 0),
// so each scan step is a (16 x 64) x (64 x 64) f32 GEMM per batch-group ->
// V_WMMA_F32_16X16X4_F32 (full f32 precision, 16 chained K=4 steps).
// 8 blocks own 16 batches each (WMMA M dim); 4 waves per block own one
// 16-column N-tile.  Problem is tiny (6.5 MB feats, ~0.4 GFLOP) and
// latency-bound on the sequential L=256 scan: hot state lives in LDS/VGPRs,
// per-step exp/log use the hardware v_exp_f32/v_log_f32 (args <= 0, so the
// precise-path overflow fixups are dead weight), per-step global loads are
// hoisted above the WMMA chain, and t+1 feats are prefetched
// (global_prefetch_b8).
//
// WMMA fragment layout (CDNA5 ISA 7.12.2, 32-bit operands):
//   A 16x4 : lanes 0-15 hold M=lane, VGPR {x,y} = K{0,1}; lanes 16-31 = K{2,3}
//   B 4x16 : lanes 0-15 hold N=lane, VGPR {x,y} = K{0,1}; lanes 16-31 = K{2,3}
//   C/D    : VGPR v, lanes 0-15 -> M=v, lanes 16-31 -> M=v+8; N = lane&15

#define B_      128
#define L_      256
#define T_      50
#define TP      64            // padded tag dimension
#define START_  48
#define STOP_   49
#define NEGINF  (-INFINITY)

typedef float v2f __attribute__((ext_vector_type(2)));
typedef float v8f __attribute__((ext_vector_type(8)));

// ---------------------------------------------------------------------------
// Gold score (deterministic single-block reduction).
// ---------------------------------------------------------------------------
__global__ void crf_gold_kernel(const float* __restrict__ feats,
                                const float* __restrict__ trans,
                                const int*   __restrict__ tags,
                                const int*   __restrict__ mask,
                                float* __restrict__ gold_out) {
  __shared__ float red[256];
  const int tid = threadIdx.x;
  float acc = 0.f;

  for (int idx = tid; idx < B_ * L_; idx += 256) {
    const int b = idx / L_;
    const int t = idx - b * L_;
    if (mask[b * L_ + t]) {
      const int tg = tags[b * L_ + t];
      const int pv = (t == 0) ? START_ : tags[b * L_ + t - 1];
      acc += feats[(b * L_ + t) * T_ + tg] + trans[pv * T_ + tg];
    }
  }
  for (int b = tid; b < B_; b += 256) {
    int len = 0;
    for (int t = 0; t < L_; ++t) len += (mask[b * L_ + t] != 0);
    const int endtag = tags[b * L_ + len - 1];
    acc += trans[endtag * T_ + STOP_];
  }

  red[tid] = acc;
  __syncthreads();
  for (int s = 128; s > 0; s >>= 1) {
    if (tid < s) red[tid] += red[tid + s];
    __syncthreads();
  }
  if (tid == 0) gold_out[0] = red[0];
}

// ---------------------------------------------------------------------------
// Forward scan: grid = 8 blocks (16 batches each), block = 128 threads
// (4 waves; wave w computes N-tile w).
// ---------------------------------------------------------------------------
__global__ void __launch_bounds__(128, 1)
crf_forward_kernel(const float* __restrict__ feats,
                   const float* __restrict__ trans,
                   const int*   __restrict__ mask,
                   float* __restrict__ partial_out) {
  __shared__ __align__(16) float part[16 * TP];        // 4 KB: log-space state
  __shared__ __align__(16) v2f   bfrag[4 * 16 * 32];   // 16 KB: E B-fragments
  __shared__ float red[16];

  const int tid  = threadIdx.x;
  const int lane = tid & 31;
  const int wave = tid >> 5;      // N-tile index 0..3
  const int blk  = blockIdx.x;    // batch group: rows 16*blk .. 16*blk+15
  const int c    = lane & 15;
  const int hi   = lane >> 4;     // 0 = lanes 0-15, 1 = lanes 16-31
  const int k0   = 2 * hi;        // K offset of this half-wave in a fragment

  // --- Build B fragments of E = exp(trans), zero-padded to 64x64 (precise
  //     expf: one-time cost, E's accuracy feeds all 255 steps) ---
  for (int idx = tid; idx < 4 * 16 * 32; idx += 128) {
    const int n   = idx >> 9;
    const int kt  = (idx >> 5) & 15;
    const int l   = idx & 31;
    const int lhi = l >> 4;
    const int lc  = l & 15;
    const int i0  = 4 * kt + 2 * lhi;
    const int j   = 16 * n + lc;
    v2f v;
    v.x = (i0     < T_ && j < T_) ? expf(trans[(i0    ) * T_ + j]) : 0.f;
    v.y = (i0 + 1 < T_ && j < T_) ? expf(trans[(i0 + 1) * T_ + j]) : 0.f;
    bfrag[idx] = v;
  }

  // --- part0[b,j] = feats[b,0,j] + trans[START,j]  (pads -> -inf) ---
  for (int idx = tid; idx < 16 * TP; idx += 128) {
    const int r = idx >> 6;
    const int j = idx & 63;
    const int b = 16 * blk + r;
    part[idx] = (j < T_) ? feats[(b * L_) * T_ + j] + trans[START_ * T_ + j]
                         : NEGINF;
  }
  __syncthreads();

  const int col = 16 * wave + c;

  // --- sequential scan over time ---
#pragma unroll 1
  for (int t = 1; t < L_; ++t) {
    // Row max of batch-row (lane&15): each half-wave maxes 32 columns,
    // combined across halves with one shfl_xor (uniform control flow).
    const float* prow = &part[c * TP];
    float m = NEGINF;
#pragma unroll
    for (int j4 = 0; j4 < 32; j4 += 4) {
      const float4 v = *(const float4*)&prow[32 * hi + j4];
      m = fmaxf(m, fmaxf(fmaxf(v.x, v.y), fmaxf(v.z, v.w)));
    }
    m = fmaxf(m, __shfl_xor(m, 16));

    // A fragments: exp(part - m), args <= 0 -> hardware v_exp_f32 is safe.
    v2f afr[16];
#pragma unroll
    for (int kt = 0; kt < 16; ++kt) {
      const v2f p = *(const v2f*)&prow[4 * kt + k0];
      v2f a;
      a.x = __expf(p.x - m);
      a.y = __expf(p.y - m);
      afr[kt] = a;
    }

    // Hoist this step's global loads above the WMMA chain so VMEM latency
    // overlaps matrix work; prefetch next step's feats row (global_prefetch).
    float fv[8];
    int   mtv[8];
#pragma unroll
    for (int v = 0; v < 8; ++v) {
      const int M = v + 8 * hi;
      const int b = 16 * blk + M;
      fv[v]  = (col < T_) ? feats[(b * L_ + t) * T_ + col] : 0.f;
      mtv[v] = mask[b * L_ + t];
      if (col < T_ && t + 1 < L_)
        __builtin_prefetch(&feats[(b * L_ + t + 1) * T_ + col], 0, 3);
    }

    // 16 chained f32 WMMAs: acc(16x16) += A(16x4) * E(4x16)
    v8f acc = {0.f, 0.f, 0.f, 0.f, 0.f, 0.f, 0.f, 0.f};
#pragma unroll
    for (int kt = 0; kt < 16; ++kt) {
      const v2f bf = bfrag[(wave * 16 + kt) * 32 + lane];
      acc = __builtin_amdgcn_wmma_f32_16x16x4_f32(
          false, afr[kt], false, bf, (short)0, acc, false, false);
    }

    __syncthreads();  // all reads of part done before this step's writes

#pragma unroll
    for (int v = 0; v < 8; ++v) {
      const int M = v + 8 * hi;
      const float mb = __shfl(m, M);        // row-max of batch row M
      float nw;
      if (col < T_) {
        nw = mb + __logf(acc[v]) + fv[v];   // log(0) -> -inf is correct
      } else {
        nw = NEGINF;                        // padded columns stay absorbing
      }
      const float old = part[M * TP + col]; // unique (M,col) owner: no race
      part[M * TP + col] = mtv[v] ? nw : old;
    }
    __syncthreads();
  }

  // --- terminal: term_b = logsumexp_i(part[b,i] + trans[i,STOP]) ---
  // Done in log space (trans[:,STOP] = -1e4 would underflow in exp-space);
  // precise expf/logf here, it runs once.
  if (tid < 16) {
    const int r = tid;
    float mr = NEGINF, s = 0.f;
    for (int i = 0; i < T_; ++i) {
      const float x = part[r * TP + i] + trans[i * T_ + STOP_];
      if (x > -3.0e38f) {                 // skip -inf absorbing entries
        if (x > mr) { s = s * expf(mr - x) + 1.f; mr = x; }
        else        { s += expf(x - mr); }
      }
    }
    red[r] = mr + logf(s);
  }
  __syncthreads();
  if (tid == 0) {
    float tot = 0.f;
    for (int i = 0; i < 16; ++i) tot += red[i];
    partial_out[blk] = tot;
  }
}

// ---------------------------------------------------------------------------
// Combine: out = sum(forward partials) - gold.  No float atomics ->
// bit-identical across graph replays.
// ---------------------------------------------------------------------------
__global__ void crf_combine_kernel(const float* __restrict__ ws,
                                   float* __restrict__ out) {
  if (threadIdx.x == 0) {
    float f = 0.f;
    for (int i = 0; i < 8; ++i) f += ws[1 + i];
    out[0] = f - ws[0];
  }
}

extern "C" void kernel_launch(void* const* d_in, const int* in_sizes, int n_in,
                              void* d_out, int out_size, void* d_ws, size_t ws_size,
                              hipStream_t stream) {
  (void)in_sizes; (void)n_in; (void)out_size; (void)ws_size;
  const float* feats = (const float*)d_in[0];
  const float* trans = (const float*)d_in[1];
  const int*   tags  = (const int*)d_in[2];
  const int*   mask  = (const int*)d_in[3];
  float* ws = (float*)d_ws;   // ws[0] = gold, ws[1..8] = forward partials

  crf_gold_kernel<<<1, 256, 0, stream>>>(feats, trans, tags, mask, ws);
  crf_forward_kernel<<<8, 128, 0, stream>>>(feats, trans, mask, ws + 1);
  crf_combine_kernel<<<1, 32, 0, stream>>>(ws, (float*)d_out);
}